// GAT_3599182594390
// MI455X (gfx1250) — compile-verified
//
#include <hip/hip_runtime.h>
#include <hip/hip_bf16.h>
#include <stdint.h>

// ---------------------------------------------------------------------------
// GAT layer for MI455X (gfx1250, wave32).
//   N=100000 nodes, E=1.6M edges, IN=256, OUT=64, H=4 heads.
//   Fused GEMM z = h @ Wflat^T (Wflat = W reshaped [H*OUT][IN]) via
//   V_WMMA_F32_16X16X4_F32 (fp32 matrix core, exact precision).
//   Edge softmax via ordered-uint atomicMax + exp + atomicAdd passes.
//   Aggregation: one wave32 per (edge, head), 2 features/lane, f32 atomics.
// ---------------------------------------------------------------------------

#define N_NODES 100000
#define N_EDGES 1600000
#define DIN     256
#define DOUT    64
#define NHEAD   4
#define DTOT    (NHEAD * DOUT)   // 256 = concat output width
#define SLOPE   0.01f

typedef __attribute__((ext_vector_type(2))) float v2f;
typedef __attribute__((ext_vector_type(8))) float v8f;

// Monotonic float <-> uint encoding so unsigned atomicMax == float max.
__device__ __forceinline__ unsigned ordEnc(float f) {
    unsigned u = __float_as_uint(f);
    return (u & 0x80000000u) ? ~u : (u | 0x80000000u);
}
__device__ __forceinline__ float ordDec(unsigned u) {
    unsigned v = (u & 0x80000000u) ? (u & 0x7fffffffu) : ~u;
    return __uint_as_float(v);
}
#define ORD_NEG_INF 0x007FFFFFu   // ordEnc(-inf)

// ---------------------------------------------------------------------------
// 0) init: zero d_out and den, set segment-max accumulators to -inf.
// ---------------------------------------------------------------------------
__global__ void gat_init(float* __restrict__ out, float* __restrict__ den,
                         unsigned* __restrict__ mord, long nOut, long nHN) {
    long i = (long)blockIdx.x * blockDim.x + threadIdx.x;
    long stride = (long)gridDim.x * blockDim.x;
    for (long k = i; k < nOut; k += stride) out[k] = 0.0f;
    for (long k = i; k < nHN; k += stride) { den[k] = 0.0f; mord[k] = ORD_NEG_INF; }
}

// ---------------------------------------------------------------------------
// 1) GEMM: z[N][256] = h[N][256] @ Wflat[256][256]^T, fp32 WMMA 16x16x4.
//    Block = 512 threads = 16 waves; block b -> M-tile b (16 rows),
//    wave w -> N-tile w (16 cols). K loop: 64 x WMMA(K=4).
//    A frag (16x4 f32): lanes 0-15 hold {K0,K1} of row M=lane,
//                       lanes 16-31 hold {K2,K3} of row M=lane-16.
//    B frag (4x16 f32): mirrored (row = Wflat column block).
// ---------------------------------------------------------------------------
__global__ __launch_bounds__(512) void gat_gemm(const float* __restrict__ h,
                                                const float* __restrict__ Wf,
                                                float* __restrict__ z) {
    const int wave = threadIdx.x >> 5;
    const int lane = threadIdx.x & 31;
    const int r    = lane & 15;
    const int koff = (lane < 16) ? 0 : 2;
    const int m0 = blockIdx.x * 16;
    const int c0 = wave * 16;

    const float* ap = h  + (size_t)(m0 + r) * DIN + koff;
    const float* bp = Wf + (size_t)(c0 + r) * DIN + koff;

    v8f acc = {};
#pragma unroll 4
    for (int k0 = 0; k0 < DIN; k0 += 4) {
        v2f a = *(const v2f*)(ap + k0);   // 8B aligned: k0%4==0, koff in {0,2}
        v2f b = *(const v2f*)(bp + k0);
        acc = __builtin_amdgcn_wmma_f32_16x16x4_f32(
            /*neg_a=*/false, a, /*neg_b=*/false, b,
            /*c_mod=*/(short)0, acc, /*reuse_a=*/false, /*reuse_b=*/false);
    }

    // C/D layout: VGPR v -> M = v + (lane<16 ? 0 : 8), N = lane&15.
    const int rbase = (lane < 16) ? 0 : 8;
    float* zp = z + (size_t)(m0 + rbase) * DTOT + c0 + r;
#pragma unroll
    for (int v = 0; v < 8; ++v) zp[(size_t)v * DTOT] = acc[v];
}

// ---------------------------------------------------------------------------
// 2) scores: s_src[h][n] = z[n, h*64:+64] . a_src ; same for s_dst.
//    Reference uses attn[0] for ALL heads (a_src=attn[0][:64], a_dst=attn[0][64:]).
// ---------------------------------------------------------------------------
__global__ void gat_score(const float* __restrict__ z, const float* __restrict__ attn,
                          float* __restrict__ ssrc, float* __restrict__ sdst) {
    int idx = blockIdx.x * blockDim.x + threadIdx.x;
    if (idx >= N_NODES * NHEAD) return;
    const int hd = idx & (NHEAD - 1);
    const int n  = idx >> 2;
    const float* zr = z + (size_t)n * DTOT + hd * DOUT;
    const float* as = attn;          // attn[0][0:64]
    const float* ad = attn + DOUT;   // attn[0][64:128]
    float s0 = 0.0f, s1 = 0.0f;
#pragma unroll 8
    for (int j = 0; j < DOUT; ++j) {
        float v = zr[j];
        s0 = fmaf(v, as[j], s0);
        s1 = fmaf(v, ad[j], s1);
    }
    ssrc[(size_t)hd * N_NODES + n] = s0;
    sdst[(size_t)hd * N_NODES + n] = s1;
}

// ---------------------------------------------------------------------------
// 3) per-edge leaky_relu score + segment_max (ordered-uint atomicMax).
// ---------------------------------------------------------------------------
__global__ void gat_edge_max(const float* __restrict__ ssrc, const float* __restrict__ sdst,
                             const int* __restrict__ esrc, const int* __restrict__ edst,
                             float* __restrict__ escr, unsigned* __restrict__ mord) {
    int tid = blockIdx.x * blockDim.x + threadIdx.x;
    if (tid >= N_EDGES * NHEAD) return;
    const int hd = tid / N_EDGES;
    const int e  = tid - hd * N_EDGES;
    const int s = esrc[e], d = edst[e];
    float x = ssrc[(size_t)hd * N_NODES + s] + sdst[(size_t)hd * N_NODES + d];
    float ee = (x > 0.0f) ? x : SLOPE * x;
    escr[(size_t)hd * N_EDGES + e] = ee;
    atomicMax(&mord[(size_t)hd * N_NODES + d], ordEnc(ee));
}

// ---------------------------------------------------------------------------
// 4) ex = exp(e - m[dst]) (m := 0 when -inf / non-finite), segment_sum -> den.
// ---------------------------------------------------------------------------
__global__ void gat_edge_exp(float* __restrict__ escr, const unsigned* __restrict__ mord,
                             const int* __restrict__ edst, float* __restrict__ den) {
    int tid = blockIdx.x * blockDim.x + threadIdx.x;
    if (tid >= N_EDGES * NHEAD) return;
    const int hd = tid / N_EDGES;
    const int e  = tid - hd * N_EDGES;
    const int d = edst[e];
    float m = ordDec(mord[(size_t)hd * N_NODES + d]);
    if (!(m >= -3.402823466e38f)) m = 0.0f;   // -inf or NaN -> 0 (matches isfinite guard)
    size_t i = (size_t)hd * N_EDGES + e;
    float ex = __expf(escr[i] - m);
    escr[i] = ex;
    atomicAdd(&den[(size_t)hd * N_NODES + d], ex);
}

// ---------------------------------------------------------------------------
// 5) alpha = ex / max(den[dst], 1e-9)  (overwrites escr in place).
// ---------------------------------------------------------------------------
__global__ void gat_edge_alpha(float* __restrict__ escr, const float* __restrict__ den,
                               const int* __restrict__ edst) {
    int tid = blockIdx.x * blockDim.x + threadIdx.x;
    if (tid >= N_EDGES * NHEAD) return;
    const int hd = tid / N_EDGES;
    const int e  = tid - hd * N_EDGES;
    const int d = edst[e];
    size_t i = (size_t)hd * N_EDGES + e;
    escr[i] = escr[i] / fmaxf(den[(size_t)hd * N_NODES + d], 1e-9f);
}

// ---------------------------------------------------------------------------
// 6) aggregate: out[dst, hd*64+j] += alpha * z[src, hd*64+j].
//    One wave32 per (edge, head); lane handles 2 consecutive features
//    (coalesced 256B gather of z row segment, f32 atomic scatter).
// ---------------------------------------------------------------------------
__global__ __launch_bounds__(256) void gat_agg(const float* __restrict__ alp,
                                               const float* __restrict__ z,
                                               const int* __restrict__ esrc,
                                               const int* __restrict__ edst,
                                               float* __restrict__ out) {
    const int gw   = blockIdx.x * 8 + (threadIdx.x >> 5);
    const int lane = threadIdx.x & 31;
    if (gw >= N_EDGES * NHEAD) return;
    const int hd = gw / N_EDGES;
    const int e  = gw - hd * N_EDGES;
    const int s = esrc[e], d = edst[e];
    const float a = alp[(size_t)hd * N_EDGES + e];
    const float2 zz = *(const float2*)(z + (size_t)s * DTOT + hd * DOUT + 2 * lane);
    float* op = out + (size_t)d * DTOT + hd * DOUT + 2 * lane;
    atomicAdd(op,     a * zz.x);
    atomicAdd(op + 1, a * zz.y);
}

// ---------------------------------------------------------------------------
extern "C" void kernel_launch(void* const* d_in, const int* in_sizes, int n_in,
                              void* d_out, int out_size, void* d_ws, size_t ws_size,
                              hipStream_t stream) {
    const float* h    = (const float*)d_in[0];   // [N][256]
    const float* W    = (const float*)d_in[1];   // [H][64][256] == Wflat[256][256]
    const float* attn = (const float*)d_in[2];   // [H][128], only row 0 used
    const int*   esrc = (const int*)d_in[3];     // [E]
    const int*   edst = (const int*)d_in[4];     // [E]
    float* out = (float*)d_out;                  // [N][256] fp32

    // Workspace carve-out (256B-aligned regions), total ~135 MB.
    char* ws = (char*)d_ws;
    size_t off = 0;
    auto carve = [&](size_t bytes) -> void* {
        void* p = ws + off;
        off += (bytes + 255) & ~(size_t)255;
        return p;
    };
    float*    z    = (float*)   carve((size_t)N_NODES * DTOT * sizeof(float));   // 102.4 MB
    float*    ssrc = (float*)   carve((size_t)NHEAD * N_NODES * sizeof(float));
    float*    sdst = (float*)   carve((size_t)NHEAD * N_NODES * sizeof(float));
    unsigned* mord = (unsigned*)carve((size_t)NHEAD * N_NODES * sizeof(unsigned));
    float*    den  = (float*)   carve((size_t)NHEAD * N_NODES * sizeof(float));
    float*    escr = (float*)   carve((size_t)NHEAD * N_EDGES * sizeof(float));  // 25.6 MB
    (void)ws_size; (void)in_sizes; (void)n_in; (void)out_size;

    const long nOut = (long)N_NODES * DTOT;
    const long nHN  = (long)NHEAD * N_NODES;
    gat_init<<<4096, 256, 0, stream>>>(out, den, mord, nOut, nHN);

    gat_gemm<<<N_NODES / 16, 512, 0, stream>>>(h, W, z);   // 6250 M-tiles, exact

    const int nScore = N_NODES * NHEAD;
    gat_score<<<(nScore + 255) / 256, 256, 0, stream>>>(z, attn, ssrc, sdst);

    const int nEH = N_EDGES * NHEAD;                       // 6.4M
    gat_edge_max  <<<(nEH + 255) / 256, 256, 0, stream>>>(ssrc, sdst, esrc, edst, escr, mord);
    gat_edge_exp  <<<(nEH + 255) / 256, 256, 0, stream>>>(escr, mord, edst, den);
    gat_edge_alpha<<<(nEH + 255) / 256, 256, 0, stream>>>(escr, den, edst);

    gat_agg<<<nEH / 8, 256, 0, stream>>>(escr, z, esrc, edst, out);  // 1 wave per (edge,head)
}